// GATModel_32770600468657
// MI455X (gfx1250) — compile-verified
//
#include <hip/hip_runtime.h>
#include <math.h>

#define HEADS 4
#define NEG_SLOPE 0.2f

typedef __attribute__((ext_vector_type(2))) float v2f;
typedef __attribute__((ext_vector_type(8))) float v8f;

static __device__ __forceinline__ void atomicMaxF(float* addr, float v) {
  // sign-aware int/uint trick; valid with -inf initialization
  if (v >= 0.0f) atomicMax(reinterpret_cast<int*>(addr), __float_as_int(v));
  else atomicMin(reinterpret_cast<unsigned int*>(addr), __float_as_uint(v));
}

// ---------------- fp32 WMMA GEMM: H = X[nrows x IND] @ W[IND x OUTD] ----------------
// One wave computes one 16x16 tile; OUTD/16 waves per block cover one 16-row stripe.
template<int IND, int OUTD>
__global__ __launch_bounds__(32 * (OUTD / 16))
void gat_gemm_wmma(const float* __restrict__ X, const float* __restrict__ W,
                   float* __restrict__ H, int nrows) {
  const int wave = threadIdx.x >> 5;   // column tile
  const int lane = threadIdx.x & 31;
  const int lr   = lane & 15;
  const int hi   = lane >> 4;          // 0: K 0,1   1: K 2,3
  const int row0 = blockIdx.x * 16;
  const int col0 = wave * 16;
  int rowA = row0 + lr;
  if (rowA >= nrows) rowA = nrows - 1;        // clamp (no divergent exit: EXEC stays all-1 for WMMA)

  v8f acc = {};
  const float* xbase = X + (size_t)rowA * IND + hi * 2;
  const float* wbase = W + (size_t)(hi * 2) * OUTD + col0 + lr;
  #pragma unroll 4
  for (int k = 0; k < IND; k += 4) {
    v2f a, b;
    a[0] = xbase[k];
    a[1] = xbase[k + 1];
    const float* wp = wbase + (size_t)k * OUTD;
    b[0] = wp[0];
    b[1] = wp[OUTD];
    acc = __builtin_amdgcn_wmma_f32_16x16x4_f32(false, a, false, b, (short)0, acc,
                                                false, false);
  }
  // C/D layout: VGPR r holds row M = r + hi*8, lanes 0..15 -> N = lr
  float* obase = H + (size_t)row0 * OUTD + col0 + lr;
  if (row0 + 16 <= nrows) {
    // uniform full-tile fast path: 8 clean stores, no per-row EXEC churn
    #pragma unroll
    for (int r = 0; r < 8; ++r) obase[(size_t)(r + hi * 8) * OUTD] = acc[r];
  } else {
    #pragma unroll
    for (int r = 0; r < 8; ++r) {
      int m = r + hi * 8;
      if (row0 + m < nrows) obase[(size_t)m * OUTD] = acc[r];
    }
  }
}

// ---------------- per-node attention logits ----------------
__global__ void logits_kernel(const float* __restrict__ H, const float* __restrict__ a_src,
                              const float* __restrict__ a_dst, float* __restrict__ al_s,
                              float* __restrict__ al_d, int n, int outD) {
  int idx = blockIdx.x * blockDim.x + threadIdx.x;  // node*HEADS + head
  if (idx >= n * HEADS) return;
  int node = idx >> 2, head = idx & 3;
  int hd = outD / HEADS;
  const float* hp = H + (size_t)node * outD + head * hd;
  const float* as = a_src + head * hd;
  const float* ad = a_dst + head * hd;
  float s = 0.f, d = 0.f;
  for (int i = 0; i < hd; ++i) { s += hp[i] * as[i]; d += hp[i] * ad[i]; }
  al_s[idx] = s;
  al_d[idx] = d;
}

// ---------------- init accumulators (every call: ws is not re-poisoned) ----------------
__global__ void init_kernel(float* __restrict__ m, float* __restrict__ denom,
                            float* __restrict__ agg, int total, int n4) {
  int i = blockIdx.x * blockDim.x + threadIdx.x;
  if (i < n4) { m[i] = -INFINITY; denom[i] = 0.f; }
  if (i < total) agg[i] = 0.f;
}

static __device__ __forceinline__ void edge_sd(int e, int E, const int* src, const int* dst,
                                               int& s, int& d) {
  if (e < E) { s = src[e]; d = dst[e]; }
  else { s = d = e - E; }            // fused self-loops
}

// ---------------- segment max over incoming edges ----------------
__global__ void edge_max_kernel(const int* __restrict__ src, const int* __restrict__ dst,
                                int E, int n, const float* __restrict__ al_s,
                                const float* __restrict__ al_d, float* __restrict__ m) {
  int idx = blockIdx.x * blockDim.x + threadIdx.x;
  if (idx >= (E + n) * HEADS) return;
  int e = idx >> 2, h = idx & 3;
  int s, d; edge_sd(e, E, src, dst, s, d);
  float v = al_s[s * HEADS + h] + al_d[d * HEADS + h];
  v = (v > 0.f) ? v : NEG_SLOPE * v;
  atomicMaxF(&m[d * HEADS + h], v);
}

// ---------------- segment sum of exp ----------------
__global__ void edge_sum_kernel(const int* __restrict__ src, const int* __restrict__ dst,
                                int E, int n, const float* __restrict__ al_s,
                                const float* __restrict__ al_d, const float* __restrict__ m,
                                float* __restrict__ denom) {
  int idx = blockIdx.x * blockDim.x + threadIdx.x;
  if (idx >= (E + n) * HEADS) return;
  int e = idx >> 2, h = idx & 3;
  int s, d; edge_sd(e, E, src, dst, s, d);
  float v = al_s[s * HEADS + h] + al_d[d * HEADS + h];
  v = (v > 0.f) ? v : NEG_SLOPE * v;
  atomicAdd(&denom[d * HEADS + h], __expf(v - m[d * HEADS + h]));
}

// ---------------- weighted message aggregation: one wave per edge ----------------
template<int OUTD>
__global__ void edge_agg_kernel(const int* __restrict__ src, const int* __restrict__ dst,
                                int E, int n, const float* __restrict__ al_s,
                                const float* __restrict__ al_d, const float* __restrict__ m,
                                const float* __restrict__ denom, const float* __restrict__ H,
                                float* __restrict__ agg) {
  int wid  = (blockIdx.x * blockDim.x + threadIdx.x) >> 5;
  int lane = threadIdx.x & 31;
  if (wid >= E + n) return;
  int s, d; edge_sd(wid, E, src, dst, s, d);
  constexpr int VEC = OUTD / 32;     // channels per lane (contiguous, same head)
  constexpr int HD  = OUTD / HEADS;
  int c0   = lane * VEC;
  int head = c0 / HD;
  float e = al_s[s * HEADS + head] + al_d[d * HEADS + head];
  e = (e > 0.f) ? e : NEG_SLOPE * e;
  float alpha = __expf(e - m[d * HEADS + head]) / (denom[d * HEADS + head] + 1e-16f);
  const float* hrow = H + (size_t)s * OUTD + c0;
  float* arow = agg + (size_t)d * OUTD + c0;
  #pragma unroll
  for (int i = 0; i < VEC; ++i) atomicAdd(&arow[i], alpha * hrow[i]);
}

// ---------------- bias + relu ----------------
__global__ void finalize_kernel(const float* __restrict__ agg, const float* __restrict__ b,
                                float* __restrict__ outF, int n, int outD) {
  int i = blockIdx.x * blockDim.x + threadIdx.x;
  if (i >= n * outD) return;
  float v = agg[i] + b[i % outD];
  outF[i] = (v > 0.f) ? v : 0.f;
}

// ---------------- classifier: 64 -> 40 ----------------
__global__ void classifier_kernel(const float* __restrict__ Hf, const float* __restrict__ Wc,
                                  const float* __restrict__ bc, float* __restrict__ out, int n) {
  int idx = blockIdx.x * blockDim.x + threadIdx.x;
  if (idx >= n * 40) return;
  int node = idx / 40, j = idx - node * 40;
  const float* hr = Hf + (size_t)node * 64;
  float acc = bc[j];
  #pragma unroll 8
  for (int k = 0; k < 64; ++k) acc += hr[k] * Wc[k * 40 + j];
  out[idx] = acc;
}

static inline int cdiv(long long a, long long b) { return (int)((a + b - 1) / b); }

template<int IND, int OUTD>
static void run_layer(const float* in, const float* W, const float* a_s, const float* a_d,
                      const float* bias, const int* esrc, const int* edst, int E, int n,
                      float* Hbuf, float* Abuf, float* outF, float* als, float* ald,
                      float* mm, float* dn, hipStream_t stream) {
  gat_gemm_wmma<IND, OUTD><<<cdiv(n, 16), 32 * (OUTD / 16), 0, stream>>>(in, W, Hbuf, n);
  int nh = n * HEADS;
  logits_kernel<<<cdiv(nh, 256), 256, 0, stream>>>(Hbuf, a_s, a_d, als, ald, n, OUTD);
  int tot = n * OUTD;
  init_kernel<<<cdiv(tot, 256), 256, 0, stream>>>(mm, dn, Abuf, tot, nh);
  int Etot = E + n;
  edge_max_kernel<<<cdiv((long long)Etot * HEADS, 256), 256, 0, stream>>>(esrc, edst, E, n,
                                                                          als, ald, mm);
  edge_sum_kernel<<<cdiv((long long)Etot * HEADS, 256), 256, 0, stream>>>(esrc, edst, E, n,
                                                                          als, ald, mm, dn);
  edge_agg_kernel<OUTD><<<cdiv((long long)Etot * 32, 256), 256, 0, stream>>>(
      esrc, edst, E, n, als, ald, mm, dn, Hbuf, Abuf);
  finalize_kernel<<<cdiv(tot, 256), 256, 0, stream>>>(Abuf, bias, outF, n, OUTD);
}

extern "C" void kernel_launch(void* const* d_in, const int* in_sizes, int n_in,
                              void* d_out, int out_size, void* d_ws, size_t ws_size,
                              hipStream_t stream) {
  const float* x    = (const float*)d_in[0];
  const int*   eix  = (const int*)d_in[1];
  const float* W0   = (const float*)d_in[2];
  const float* as0  = (const float*)d_in[3];
  const float* ad0  = (const float*)d_in[4];
  const float* b0   = (const float*)d_in[5];
  const float* W1   = (const float*)d_in[6];
  const float* as1  = (const float*)d_in[7];
  const float* ad1  = (const float*)d_in[8];
  const float* b1   = (const float*)d_in[9];
  const float* W2   = (const float*)d_in[10];
  const float* as2  = (const float*)d_in[11];
  const float* ad2  = (const float*)d_in[12];
  const float* b2   = (const float*)d_in[13];
  const float* Wc   = (const float*)d_in[14];
  const float* bc   = (const float*)d_in[15];

  const int n = in_sizes[0] / 128;        // N nodes
  const int E = in_sizes[1] / 2;          // directed edges (before self-loops)
  const int* esrc = eix;
  const int* edst = eix + E;

  // workspace carve-up (floats)
  float* F0  = (float*)d_ws;              // GEMM output h       [n x 128]
  float* F1  = F0 + (size_t)n * 128;      // aggregation buffer  [n x 128]
  float* F2  = F1 + (size_t)n * 128;      // layer features      [n x 128]
  float* als = F2 + (size_t)n * 128;      // [n x 4]
  float* ald = als + (size_t)n * 4;
  float* mm  = ald + (size_t)n * 4;
  float* dn  = mm + (size_t)n * 4;

  // layer 0: x(128) -> F2(128)
  run_layer<128, 128>(x,  W0, as0, ad0, b0, esrc, edst, E, n, F0, F1, F2, als, ald, mm, dn, stream);
  // layer 1: F2(128) -> F2(128)   (F2 is free once the GEMM has consumed it)
  run_layer<128, 128>(F2, W1, as1, ad1, b1, esrc, edst, E, n, F0, F1, F2, als, ald, mm, dn, stream);
  // layer 2: F2(128) -> F2(64)
  run_layer<128, 64>(F2, W2, as2, ad2, b2, esrc, edst, E, n, F0, F1, F2, als, ald, mm, dn, stream);

  classifier_kernel<<<cdiv((long long)n * 40, 256), 256, 0, stream>>>(F2, Wc, bc,
                                                                      (float*)d_out, n);
}